// GradReBlur_30777735643683
// MI455X (gfx1250) — compile-verified
//
#include <hip/hip_runtime.h>
#include <hip/hip_bf16.h>
#include <stdint.h>

#define BATCH 16
#define CHAN  3
#define HH    256
#define WW    256
#define NF    9
#define TILE  32
#define HALO  8
#define SW    (TILE + 2*HALO)        // 48: staged gradient window
#define PLANE (SW*SW)                // 2304 cells (float2 each)
#define BW    (TILE + 2)             // 34: staged blur window (+1 halo)
#define BPLANE (BW*BW)               // 1156
#define GRAD_CELLS (CHAN*PLANE)      // 6912 float2 cells
#define BLUR_CELLS (CHAN*BPLANE)     // 3468 float cells
#define TPB   256
#define NBLK2 (BATCH * 64)           // 1024 blocks: 16 batches x 8x8 tiles
#define TOTAL_ELEMS (BATCH*CHAN*HH*WW)  // 3145728

__device__ __forceinline__ float ssqrt(float v) {
  return v > 0.f ? __builtin_sqrtf(v) : 0.f;
}

// Branch-free zero-padded load: clamp address, always load, cndmask result.
__device__ __forceinline__ float img_ld(const float* __restrict__ p, int y, int x) {
  int yc = min(max(y, 0), HH - 1);
  int xc = min(max(x, 0), WW - 1);
  float v = p[(yc << 8) + xc];
  return ((unsigned)y < (unsigned)HH && (unsigned)x < (unsigned)WW) ? v : 0.f;
}

// ---------------------------------------------------------------------------
// Kernel 1: sharp -> interleaved {sqrt(relu(sobel_x)), sqrt(relu(sobel_y))}
// Cross-correlation (XLA conv semantics), SAME zero padding.
// ---------------------------------------------------------------------------
__global__ __launch_bounds__(TPB) void sobel_kernel(
    const float* __restrict__ in, float2* __restrict__ gxy) {
  int idx = blockIdx.x * TPB + threadIdx.x;   // over B*C*H*W
  int x  = idx & (WW - 1);
  int y  = (idx >> 8) & (HH - 1);
  int bc = idx >> 16;
  const float* p = in + ((size_t)bc << 16);
  float a  = img_ld(p, y-1, x-1), bv = img_ld(p, y-1, x), cv = img_ld(p, y-1, x+1);
  float d  = img_ld(p, y,   x-1),                          f  = img_ld(p, y,   x+1);
  float g  = img_ld(p, y+1, x-1), h  = img_ld(p, y+1, x), i  = img_ld(p, y+1, x+1);
  float vx = (a - cv) + 2.f*(d - f) + (g - i);            // SOBEL_X correlation
  float vy = (a + 2.f*bv + cv) - (g + 2.f*h + i);         // SOBEL_Y correlation
  gxy[idx] = make_float2(ssqrt(vx), ssqrt(vy));           // single b64 store
}

// ---------------------------------------------------------------------------
// Kernel 2: async-stage interleaved gradient tile (b64) + blur tile (b32)
// into LDS, bilinear re-blur from LDS (ds_load_b64 taps), squared error vs
// blur gradients from LDS, deterministic block tree reduction.
// ---------------------------------------------------------------------------
__global__ __launch_bounds__(TPB) void reblur_kernel(
    const float* __restrict__ blur, const float* __restrict__ kern,
    const float* __restrict__ wgt,  const float2* __restrict__ sXY,
    float* __restrict__ partials) {
  __shared__ float2 gsm[GRAD_CELLS];   // 3ch x 48x48 x {gx,gy}  = 55296 B
  __shared__ float  bsm[BLUR_CELLS];   // 3ch x 34x34 blur       = 13872 B
  __shared__ float  red[TPB];

  const int blk = blockIdx.x;
  const int b   = blk >> 6;             // batch
  const int t   = blk & 63;             // tile within image
  const int ty0 = (t >> 3) * TILE;
  const int tx0 = (t & 7)  * TILE;
  const int gy0 = ty0 - HALO;
  const int gx0 = tx0 - HALO;
  const int tid = threadIdx.x;

  // ---- Stage via CDNA5 async global->LDS loads (ASYNCcnt-tracked).
  // Out-of-image cells are zero-filled, reproducing the reference's
  // `valid` masking for clipped taps and SAME zero padding.
  for (int cell = tid; cell < GRAD_CELLS; cell += TPB) {   // gradient: b64
    int c   = cell / PLANE;
    int rc  = cell - c * PLANE;
    int r   = rc / SW;
    int col = rc - r * SW;
    int gy  = gy0 + r, gx = gx0 + col;
    if ((unsigned)gy < (unsigned)HH && (unsigned)gx < (unsigned)WW) {
      const float2* src = sXY + (((size_t)(b * CHAN + c)) << 16);
      uint64_t ga = (uint64_t)(uintptr_t)(src + ((gy << 8) + gx));
      uint32_t la = (uint32_t)(uintptr_t)(&gsm[cell]);   // 8B-aligned LDS off
      asm volatile("global_load_async_to_lds_b64 %0, %1, off"
                   :: "v"(la), "v"(ga) : "memory");
    } else {
      gsm[cell] = make_float2(0.f, 0.f);
    }
  }
  for (int cell = tid; cell < BLUR_CELLS; cell += TPB) {   // blur: b32
    int c   = cell / BPLANE;
    int rc  = cell - c * BPLANE;
    int r   = rc / BW;
    int col = rc - r * BW;
    int gy  = ty0 - 1 + r, gx = tx0 - 1 + col;
    if ((unsigned)gy < (unsigned)HH && (unsigned)gx < (unsigned)WW) {
      const float* src = blur + (((size_t)(b * CHAN + c)) << 16);
      uint64_t ga = (uint64_t)(uintptr_t)(src + ((gy << 8) + gx));
      uint32_t la = (uint32_t)(uintptr_t)(&bsm[cell]);
      asm volatile("global_load_async_to_lds_b32 %0, %1, off"
                   :: "v"(la), "v"(ga) : "memory");
    } else {
      bsm[cell] = 0.f;
    }
  }
  asm volatile("s_wait_asynccnt 0x0" ::: "memory");
  __syncthreads();

  float acc = 0.f;

  for (int k = 0; k < 4; ++k) {
    int pix = tid + k * TPB;             // 1024 pixels per tile
    int ly  = pix >> 5;
    int lx  = pix & 31;
    int y = ty0 + ly;
    int x = tx0 + lx;
    int yx = (y << 8) + x;

    float rbx0=0.f, rbx1=0.f, rbx2=0.f, rby0=0.f, rby1=0.f, rby2=0.f;

    for (int n = 0; n < NF; ++n) {
      // stream-once data: non-temporal loads keep L2 for reused planes
      float kx = __builtin_nontemporal_load(
          &kern[(((size_t)(b * 2*NF + n))      << 16) + yx]);
      float ky = __builtin_nontemporal_load(
          &kern[(((size_t)(b * 2*NF + NF + n)) << 16) + yx]);
      float wn = __builtin_nontemporal_load(
          &wgt [(((size_t)(b * NF + n))        << 16) + yx]);
      float px = (float)x + kx;
      float py = (float)y + ky;
      float fx = floorf(px), fy = floorf(py);
      float wx = px - fx,    wy = py - fy;
      int   xi = (int)fx,    yi = (int)fy;

      float w00 = (1.f - wx) * (1.f - wy) * wn;
      float w10 = wx * (1.f - wy) * wn;
      float w01 = (1.f - wx) * wy * wn;
      float w11 = wx * wy * wn;

      #define TAP(TX, TY, WT) do {                                            \
        int tlx = (TX) - gx0, tly = (TY) - gy0;                               \
        if ((unsigned)tlx < (unsigned)SW && (unsigned)tly < (unsigned)SW) {   \
          int lo = tly * SW + tlx;                                            \
          float2 v0 = gsm[0*PLANE + lo];                                      \
          float2 v1 = gsm[1*PLANE + lo];                                      \
          float2 v2 = gsm[2*PLANE + lo];                                      \
          rbx0 += v0.x * (WT); rby0 += v0.y * (WT);                           \
          rbx1 += v1.x * (WT); rby1 += v1.y * (WT);                           \
          rbx2 += v2.x * (WT); rby2 += v2.y * (WT);                           \
        } else if ((unsigned)(TX) < (unsigned)WW &&                           \
                   (unsigned)(TY) < (unsigned)HH) {                           \
          size_t gb = (((size_t)b * CHAN) << 16) + ((TY) << 8) + (TX);        \
          float2 v0 = sXY[gb];                                                \
          float2 v1 = sXY[gb + (1 << 16)];                                    \
          float2 v2 = sXY[gb + (2 << 16)];                                    \
          rbx0 += v0.x * (WT); rby0 += v0.y * (WT);                           \
          rbx1 += v1.x * (WT); rby1 += v1.y * (WT);                           \
          rbx2 += v2.x * (WT); rby2 += v2.y * (WT);                           \
        } } while (0)

      TAP(xi,     yi,     w00);
      TAP(xi + 1, yi,     w10);
      TAP(xi,     yi + 1, w01);
      TAP(xi + 1, yi + 1, w11);
      #undef TAP
    }

    // blur-side Sobel gradients from the staged LDS tile (zero-padded halo)
    const float rbx[3] = {rbx0, rbx1, rbx2};
    const float rby[3] = {rby0, rby1, rby2};
    int lyb = ly + 1, lxb = lx + 1;     // position inside 34x34 window
    #pragma unroll
    for (int c = 0; c < CHAN; ++c) {
      const float* bp = &bsm[c * BPLANE];
      float a  = bp[(lyb-1)*BW + (lxb-1)];
      float bv = bp[(lyb-1)*BW +  lxb   ];
      float cv = bp[(lyb-1)*BW + (lxb+1)];
      float d  = bp[ lyb   *BW + (lxb-1)];
      float f2 = bp[ lyb   *BW + (lxb+1)];
      float g  = bp[(lyb+1)*BW + (lxb-1)];
      float h2 = bp[(lyb+1)*BW +  lxb   ];
      float i2 = bp[(lyb+1)*BW + (lxb+1)];
      float bx = ssqrt((a - cv) + 2.f*(d - f2) + (g - i2));
      float by = ssqrt((a + 2.f*bv + cv) - (g + 2.f*h2 + i2));
      float ex = rbx[c] - bx;
      float ey = rby[c] - by;
      acc += ex*ex + ey*ey;
    }
  }

  // deterministic block tree reduction
  red[tid] = acc;
  __syncthreads();
  for (int s = TPB / 2; s > 0; s >>= 1) {
    if (tid < s) red[tid] += red[tid + s];
    __syncthreads();
  }
  if (tid == 0) partials[blk] = red[0];
}

// ---------------------------------------------------------------------------
// Kernel 3: deterministic final reduction of 1024 partials -> scalar
// ---------------------------------------------------------------------------
__global__ __launch_bounds__(TPB) void final_reduce(
    const float* __restrict__ partials, float* __restrict__ out) {
  __shared__ float red[TPB];
  int tid = threadIdx.x;
  float s = partials[tid] + partials[tid + 256] +
            partials[tid + 512] + partials[tid + 768];
  red[tid] = s;
  __syncthreads();
  for (int st = TPB / 2; st > 0; st >>= 1) {
    if (tid < st) red[tid] += red[tid + st];
    __syncthreads();
  }
  if (tid == 0) out[0] = red[0] * (1.0f / (float)TOTAL_ELEMS);
}

extern "C" void kernel_launch(void* const* d_in, const int* in_sizes, int n_in,
                              void* d_out, int out_size, void* d_ws, size_t ws_size,
                              hipStream_t stream) {
  (void)in_sizes; (void)n_in; (void)out_size; (void)ws_size;
  const float* blur  = (const float*)d_in[0];
  const float* sharp = (const float*)d_in[1];
  const float* kern  = (const float*)d_in[2];
  const float* wgt   = (const float*)d_in[3];

  float*  ws       = (float*)d_ws;
  float2* sXY      = (float2*)ws;                       // 3,145,728 float2
  float*  partials = ws + 2 * (size_t)TOTAL_ELEMS;      // 1,024 floats

  sobel_kernel<<<TOTAL_ELEMS / TPB, TPB, 0, stream>>>(sharp, sXY);
  reblur_kernel<<<NBLK2, TPB, 0, stream>>>(blur, kern, wgt, sXY, partials);
  final_reduce<<<1, TPB, 0, stream>>>(partials, (float*)d_out);
}